// Attention_16252156248242
// MI455X (gfx1250) — compile-verified
//
#include <hip/hip_runtime.h>
#include <cstdint>
#include <cstddef>

// ---------------------------------------------------------------------------
// DSA sparse-attention forward for MI455X (gfx1250), fp32 WMMA everywhere.
// Pipeline: projections (LDS-staged WMMA GEMM) -> RoPE -> FWHT ->
//           indexer scores (WMMA) -> exact radix top-512 ->
//           gathered flash attention (WMMA) -> output GEMM.
// ---------------------------------------------------------------------------

typedef float v2f __attribute__((ext_vector_type(2)));
typedef float v8f __attribute__((ext_vector_type(8)));

constexpr int SEQL  = 2048;
constexpr int DIMM  = 2048;
constexpr int NH    = 16;
constexpr int HD    = 128;
constexpr int IDXH  = 4;
constexpr int IDXD  = 128;
constexpr int KSEL  = 512;
constexpr float NEGV = -1.0e30f;
constexpr float ISQRT128 = 0.08838834764831845f;   // 128^-0.5

// wave-private LDS ordering: DS ops from one wave complete in order; the wait
// plus the memory clobber stops the compiler from reordering around it.
#define WAVE_LDS_FENCE() asm volatile("s_wait_dscnt 0" ::: "memory")

__device__ __forceinline__ v8f wmma_f32(v2f a, v2f b, v8f c) {
  // D = A(16x4,f32) * B(4x16,f32) + C(16x16,f32)
  return __builtin_amdgcn_wmma_f32_16x16x4_f32(false, a, false, b, (short)0, c,
                                               false, false);
}

// ---------------------------------------------------------------------------
// Row-major GEMM: C[M,N] = A[M,K] * B[K,N].  grid = (N/128, M/128), block 256.
// Block tile 128x128; each of 8 waves owns a 16x128 strip (8 accumulators).
// B is staged per 16-K chunk into LDS, TRANSPOSED (so fragments are ds b64),
// double-buffered; next chunk's global loads + A register fragments are
// prefetched before the 32-WMMA burst of the current chunk.
// ---------------------------------------------------------------------------
__global__ __launch_bounds__(256) void gemm_wmma(const float* __restrict__ A,
                                                 const float* __restrict__ B,
                                                 float* __restrict__ C,
                                                 int M, int N, int K) {
  __shared__ float bs[2][128][18];   // [buf][n_local][k_local], pad->18 (even)

  const int tid  = threadIdx.x;
  const int wave = tid >> 5;
  const int lane = tid & 31;
  const int l16  = lane & 15;        // A fragment row / B fragment col
  const int kh   = lane >> 4;        // K-half (lanes 16-31 hold K+2,K+3)
  const int m0 = blockIdx.y * 128 + wave * 16;
  const int n0 = blockIdx.x * 128;

  // cooperative B loader: 16 rows x 128 cols per chunk; thread -> 8 floats.
  const int krow_ld = tid >> 4;         // 0..15  (k_local)
  const int ncol_ld = (tid & 15) * 8;   // 0..120 (n_local)

  const float* arow = A + (size_t)(m0 + l16) * K;

  // ---- prologue: stage chunk 0 ----
  {
    const float* bp = B + (size_t)krow_ld * N + n0 + ncol_ld;
    float4 b0 = *(const float4*)(bp);
    float4 b1 = *(const float4*)(bp + 4);
    bs[0][ncol_ld + 0][krow_ld] = b0.x;
    bs[0][ncol_ld + 1][krow_ld] = b0.y;
    bs[0][ncol_ld + 2][krow_ld] = b0.z;
    bs[0][ncol_ld + 3][krow_ld] = b0.w;
    bs[0][ncol_ld + 4][krow_ld] = b1.x;
    bs[0][ncol_ld + 5][krow_ld] = b1.y;
    bs[0][ncol_ld + 6][krow_ld] = b1.z;
    bs[0][ncol_ld + 7][krow_ld] = b1.w;
  }
  v2f a_cur[4];
#pragma unroll
  for (int kk = 0; kk < 4; ++kk)
    a_cur[kk] = *(const v2f*)(arow + kk * 4 + kh * 2);
  __syncthreads();

  v8f acc[8] = {};
  for (int k0 = 0; k0 < K; k0 += 16) {
    const int buf = (k0 >> 4) & 1;

    // prefetch chunk k0+16 (global->LDS other buffer, A->registers)
    v2f a_nxt[4];
#pragma unroll
    for (int kk = 0; kk < 4; ++kk) a_nxt[kk] = a_cur[kk];
    if (k0 + 16 < K) {
      const float* bp = B + (size_t)(k0 + 16 + krow_ld) * N + n0 + ncol_ld;
      float4 b0 = *(const float4*)(bp);
      float4 b1 = *(const float4*)(bp + 4);
      bs[buf ^ 1][ncol_ld + 0][krow_ld] = b0.x;
      bs[buf ^ 1][ncol_ld + 1][krow_ld] = b0.y;
      bs[buf ^ 1][ncol_ld + 2][krow_ld] = b0.z;
      bs[buf ^ 1][ncol_ld + 3][krow_ld] = b0.w;
      bs[buf ^ 1][ncol_ld + 4][krow_ld] = b1.x;
      bs[buf ^ 1][ncol_ld + 5][krow_ld] = b1.y;
      bs[buf ^ 1][ncol_ld + 6][krow_ld] = b1.z;
      bs[buf ^ 1][ncol_ld + 7][krow_ld] = b1.w;
#pragma unroll
      for (int kk = 0; kk < 4; ++kk)
        a_nxt[kk] = *(const v2f*)(arow + k0 + 16 + kk * 4 + kh * 2);
    }

    // 32-WMMA burst on the current chunk (B fragments via aligned ds b64)
#pragma unroll
    for (int kk = 0; kk < 4; ++kk) {
#pragma unroll
      for (int nt = 0; nt < 8; ++nt) {
        v2f b = *(const v2f*)&bs[buf][nt * 16 + l16][kk * 4 + kh * 2];
        acc[nt] = wmma_f32(a_cur[kk], b, acc[nt]);
      }
    }
#pragma unroll
    for (int kk = 0; kk < 4; ++kk) a_cur[kk] = a_nxt[kk];
    __syncthreads();
  }

#pragma unroll
  for (int nt = 0; nt < 8; ++nt)
#pragma unroll
    for (int r = 0; r < 8; ++r)
      C[(size_t)(m0 + r + 8 * kh) * N + n0 + nt * 16 + l16] = acc[nt][r];
}

// ---------------------------------------------------------------------------
// Gate projection: gate[t,h] = sum_d x[t,d] * wg[d,h]   (2048x4 outputs).
// One wave per output: lanes stride the dot product (coalesced x reads),
// then a 5-step shuffle reduction.  grid = SEQ*IDXH/8, block = 256 (8 waves).
// ---------------------------------------------------------------------------
__global__ __launch_bounds__(256) void gate_kernel(const float* __restrict__ x,
                                                   const float* __restrict__ wg,
                                                   float* __restrict__ gate) {
  const int wave = threadIdx.x >> 5;
  const int lane = threadIdx.x & 31;
  const int gid = blockIdx.x * 8 + wave;   // (t,h) output id
  const int t = gid >> 2, h = gid & 3;
  const float* xr = x + (size_t)t * DIMM;
  float acc = 0.f;
#pragma unroll 8
  for (int d = lane; d < DIMM; d += 32)
    acc += xr[d] * wg[d * IDXH + h];
#pragma unroll
  for (int off = 16; off; off >>= 1)
    acc += __shfl_xor(acc, off, 32);
  if (lane == 0) gate[gid] = acc;
}

// ---------------------------------------------------------------------------
// Interleaved-pair RoPE on last 64 dims of every q head (h=0..15) and k (h=16).
// ---------------------------------------------------------------------------
__global__ void rope_kernel(float* __restrict__ qall, float* __restrict__ kmat) {
  int gid = blockIdx.x * blockDim.x + threadIdx.x;
  if (gid >= SEQL * 17 * 32) return;
  int p = gid & 31;
  int rest = gid >> 5;
  int h = rest % 17;
  int t = rest / 17;
  float inv = powf(10000.0f, -(float)p * (1.0f / 32.0f));
  float ang = (float)t * inv;
  float s, c;
  sincosf(ang, &s, &c);
  float* base = (h < 16)
      ? (qall + (size_t)t * (NH * HD) + h * HD + 64 + 2 * p)
      : (kmat + (size_t)t * HD + 64 + 2 * p);
  float u0 = base[0], u1 = base[1];
  base[0] = u0 * c - u1 * s;
  base[1] = u0 * s + u1 * c;
}

// ---------------------------------------------------------------------------
// In-place FWHT over rows of 128 elements, scaled by 128^-0.5.
// grid = nrows, block = 128.
// ---------------------------------------------------------------------------
__global__ void fwht128(float* __restrict__ data) {
  __shared__ float buf[128];
  float* p = data + (size_t)blockIdx.x * 128;
  int i = threadIdx.x;
  buf[i] = p[i];
  __syncthreads();
#pragma unroll
  for (int h = 1; h < 128; h <<= 1) {
    float a = buf[i], b = buf[i ^ h];
    __syncthreads();
    buf[i] = (i & h) ? (b - a) : (a + b);
    __syncthreads();
  }
  p[i] = buf[i] * ISQRT128;
}

// ---------------------------------------------------------------------------
// Indexer scores: scores[t,s] = 128^-0.5 * sum_h gate[t,h]*relu(qi[t,h,:].ki[s,:])
// causal-masked.  One wave per 16x16 (t,s) tile; strictly-upper tiles skipped.
// Register double-buffered 16-K chunks -> clustered loads, WMMA bursts.
// grid = (SEQ/16, SEQ/128), block = 256.
// ---------------------------------------------------------------------------
__global__ __launch_bounds__(256) void indexer_scores(
    const float* __restrict__ qi, const float* __restrict__ ki,
    const float* __restrict__ gate, float* __restrict__ scores) {
  const int wave = threadIdx.x >> 5;
  const int lane = threadIdx.x & 31;
  const int l16 = lane & 15;
  const int kh = lane >> 4;
  const int t0 = (blockIdx.y * 8 + wave) * 16;
  const int s0 = blockIdx.x * 16;
  if (s0 > t0 + 15) return;  // fully above the causal diagonal

  v8f sc = {};
  for (int h = 0; h < IDXH; ++h) {
    const float* qrow = qi + (size_t)(t0 + l16) * (IDXH * IDXD) + h * IDXD;
    const float* krow = ki + (size_t)(s0 + l16) * IDXD;

    v8f acc = {};
    v2f a_c[4], b_c[4];
#pragma unroll
    for (int kk = 0; kk < 4; ++kk) {
      a_c[kk] = *(const v2f*)(qrow + kk * 4 + kh * 2);
      b_c[kk] = *(const v2f*)(krow + kk * 4 + kh * 2);
    }
#pragma unroll
    for (int k0 = 0; k0 < IDXD; k0 += 16) {
      v2f a_n[4], b_n[4];
#pragma unroll
      for (int kk = 0; kk < 4; ++kk) { a_n[kk] = a_c[kk]; b_n[kk] = b_c[kk]; }
      if (k0 + 16 < IDXD) {
#pragma unroll
        for (int kk = 0; kk < 4; ++kk) {
          a_n[kk] = *(const v2f*)(qrow + k0 + 16 + kk * 4 + kh * 2);
          b_n[kk] = *(const v2f*)(krow + k0 + 16 + kk * 4 + kh * 2);
        }
      }
#pragma unroll
      for (int kk = 0; kk < 4; ++kk) acc = wmma_f32(a_c[kk], b_c[kk], acc);
#pragma unroll
      for (int kk = 0; kk < 4; ++kk) { a_c[kk] = a_n[kk]; b_c[kk] = b_n[kk]; }
    }
#pragma unroll
    for (int r = 0; r < 8; ++r) {
      float g = gate[(size_t)(t0 + r + 8 * kh) * IDXH + h];
      float lv = acc[r];
      sc[r] += g * (lv > 0.f ? lv : 0.f);
    }
  }
#pragma unroll
  for (int r = 0; r < 8; ++r) {
    int t = t0 + r + 8 * kh;
    int s = s0 + l16;
    float v = sc[r] * ISQRT128;
    if (s > t) v = NEGV;
    scores[(size_t)t * SEQL + s] = v;
  }
}

// ---------------------------------------------------------------------------
// Exact top-512 per row via 4-pass radix select on order-preserving uint keys,
// with deterministic prefix-scan index collection.  grid = SEQ, block = 256.
// ---------------------------------------------------------------------------
__device__ __forceinline__ unsigned f2key(float f) {
  unsigned u = __float_as_uint(f);
  return (u & 0x80000000u) ? ~u : (u | 0x80000000u);  // larger float -> larger key
}

__global__ __launch_bounds__(256) void topk_rows(const float* __restrict__ scores,
                                                 int* __restrict__ idx,
                                                 int* __restrict__ cnt) {
  const int t = blockIdx.x;
  const int tid = threadIdx.x;
  const float* srow = scores + (size_t)t * SEQL;
  int* irow = idx + (size_t)t * KSEL;

  if (t < KSEL) {  // all causal keys fit; fill rest with safe index 0
    for (int j = tid; j < KSEL; j += 256) irow[j] = (j <= t) ? j : 0;
    if (tid == 0) cnt[t] = t + 1;
    return;
  }

  __shared__ unsigned hist[256];
  __shared__ int cg[256], ce[256];
  __shared__ unsigned sh_prefix;
  __shared__ int sh_need;

  const int n = t + 1;
  unsigned prefix = 0;
  int need = KSEL;

  for (int pass = 0; pass < 4; ++pass) {
    const int shift = 24 - pass * 8;
    const unsigned mask_hi = (pass == 0) ? 0u : (0xFFFFFFFFu << (shift + 8));
    hist[tid] = 0;
    __syncthreads();
    for (int s = tid; s < n; s += 256) {
      unsigned key = f2key(srow[s]);
      if ((key & mask_hi) == prefix) atomicAdd(&hist[(key >> shift) & 0xFF], 1u);
    }
    __syncthreads();
    if (tid == 0) {
      int k = need, b = 0;
      for (int bin = 255; bin >= 0; --bin) {
        int h = (int)hist[bin];
        if (k <= h) { b = bin; break; }
        k -= h;
      }
      sh_prefix = prefix | ((unsigned)b << shift);
      sh_need = k;
    }
    __syncthreads();
    prefix = sh_prefix;
    need = sh_need;
    __syncthreads();
  }

  const unsigned T = prefix;       // threshold key (512th largest)
  const int ngt = KSEL - need;     // count strictly greater than threshold

  int lg = 0, le = 0;
  for (int s = tid; s < n; s += 256) {
    unsigned key = f2key(srow[s]);
    if (key > T) ++lg; else if (key == T) ++le;
  }
  cg[tid] = lg; ce[tid] = le;
  __syncthreads();
  if (tid == 0) {
    int ag = 0, ae = 0;
    for (int i = 0; i < 256; ++i) {
      int g = cg[i]; cg[i] = ag; ag += g;
      int e = ce[i]; ce[i] = ae; ae += e;
    }
  }
  __syncthreads();
  int pg = cg[tid], pe = ce[tid];
  for (int s = tid; s < n; s += 256) {
    unsigned key = f2key(srow[s]);
    if (key > T) {
      irow[pg++] = s;
    } else if (key == T) {
      if (pe < need) irow[ngt + pe] = s;
      ++pe;
    }
  }
  if (tid == 0) cnt[t] = KSEL;
}

// ---------------------------------------------------------------------------
// Gathered attention, one wave per query t, online softmax over 16-key tiles.
// A = q[t] (16 heads x 128), B = gathered k/v columns.  grid = SEQ/8, block=256.
// Loads for each WMMA burst are batched ahead of the burst.
// ---------------------------------------------------------------------------
__global__ __launch_bounds__(256) void attention_kernel(
    const float* __restrict__ qall, const float* __restrict__ kmat,
    const float* __restrict__ vmat, const int* __restrict__ idx,
    const int* __restrict__ cnt, float* __restrict__ omat) {
  __shared__ float sbuf_all[8][256];
  __shared__ int ibuf_all[8][16];
  __shared__ float srow_all[8][16];

  const int wave = threadIdx.x >> 5;
  const int lane = threadIdx.x & 31;
  const int l16 = lane & 15;
  const int kh = lane >> 4;
  const int t = blockIdx.x * 8 + wave;

  float* sb = sbuf_all[wave];
  int* ib = ibuf_all[wave];
  float* sr = srow_all[wave];

  const int m = cnt[t];
  const int ntile = (m + 15) >> 4;
  const int* idrow = idx + (size_t)t * KSEL;
  const float* qrow = qall + (size_t)t * (NH * HD);

  // Q A-fragments are tile-invariant: preload (row = head = l16).
  v2f aq[8];
#pragma unroll
  for (int kk = 0; kk < 8; ++kk)
    aq[kk] = *(const v2f*)(qrow + l16 * HD + kk * 4 + kh * 2);

  v8f o[8] = {};
  float mreg = -3.0e38f, lreg = 0.f;  // live in lanes 0..15 (one head-row each)

  for (int kt = 0; kt < ntile; ++kt) {
    if (lane < 16) ib[lane] = idrow[kt * 16 + lane];
    WAVE_LDS_FENCE();
    const int myidx = ib[l16];

    // QK^T tile: batch all 8 B fragments, then 8 WMMAs back-to-back.
    const float* krow = kmat + (size_t)myidx * HD;
    v2f bq[8];
#pragma unroll
    for (int kk = 0; kk < 8; ++kk)
      bq[kk] = *(const v2f*)(krow + kk * 4 + kh * 2);
    v8f att = {};
#pragma unroll
    for (int kk = 0; kk < 8; ++kk) att = wmma_f32(aq[kk], bq[kk], att);

    // scale + validity mask (this lane's 8 elements share one key column)
    const bool colvalid = (kt * 16 + l16) < m;
#pragma unroll
    for (int r = 0; r < 8; ++r) {
      float vv = colvalid ? att[r] * ISQRT128 : NEGV;
      sb[(r + 8 * kh) * 16 + l16] = vv;
    }
    WAVE_LDS_FENCE();

    // online softmax update: lanes 0..15 own head-row l16
    if (lane < 16) {
      float mx = -3.0e38f;
      for (int c2 = 0; c2 < 16; ++c2) mx = fmaxf(mx, sb[lane * 16 + c2]);
      float mnew = fmaxf(mreg, mx);
      float scl = expf(mreg - mnew);
      float ssum = 0.f;
      for (int c2 = 0; c2 < 16; ++c2) {
        float pv = expf(sb[lane * 16 + c2] - mnew);
        sb[lane * 16 + c2] = pv;
        ssum += pv;
      }
      lreg = lreg * scl + ssum;
      mreg = mnew;
      sr[lane] = scl;
    }
    WAVE_LDS_FENCE();

    // rescale output accumulators by per-row correction factor
#pragma unroll
    for (int r = 0; r < 8; ++r) {
      float f = sr[r + 8 * kh];
#pragma unroll
      for (int nt = 0; nt < 8; ++nt) o[nt][r] *= f;
    }

    // PV: o(16 heads x 128 dims) += P(16x16) * Vsel(16 x 128).
    // Per K-step: batch the P fragment + all 8 V fragments, then 8 WMMAs.
#pragma unroll
    for (int kk = 0; kk < 4; ++kk) {
      const int ka = kk * 4 + kh * 2;
      v2f a = *(const v2f*)&sb[l16 * 16 + ka];
      const float* v0 = vmat + (size_t)ib[ka] * HD;
      const float* v1 = vmat + (size_t)ib[ka + 1] * HD;
      v2f bv[8];
#pragma unroll
      for (int nt = 0; nt < 8; ++nt) {
        bv[nt].x = v0[nt * 16 + l16];
        bv[nt].y = v1[nt * 16 + l16];
      }
#pragma unroll
      for (int nt = 0; nt < 8; ++nt) o[nt] = wmma_f32(a, bv[nt], o[nt]);
    }
    WAVE_LDS_FENCE();
  }

  if (lane < 16) sr[lane] = 1.0f / lreg;
  WAVE_LDS_FENCE();

  float* orow = omat + (size_t)t * (NH * HD);
#pragma unroll
  for (int r = 0; r < 8; ++r) {
    const int hd = r + 8 * kh;
    const float invl = sr[hd];
#pragma unroll
    for (int nt = 0; nt < 8; ++nt)
      orow[hd * HD + nt * 16 + l16] = o[nt][r] * invl;
  }
}

// ---------------------------------------------------------------------------
// Launch: all stages serialized on `stream`.  Workspace layout (~61.9 MB f32):
//   q_all | k | v | qi | ki | gate | scores | o | idx(int) | cnt(int)
// ---------------------------------------------------------------------------
extern "C" void kernel_launch(void* const* d_in, const int* in_sizes, int n_in,
                              void* d_out, int out_size, void* d_ws,
                              size_t ws_size, hipStream_t stream) {
  (void)in_sizes; (void)n_in; (void)out_size; (void)ws_size;
  const float* x   = (const float*)d_in[0];
  const float* wq  = (const float*)d_in[1];
  const float* wk  = (const float*)d_in[2];
  const float* wv  = (const float*)d_in[3];
  const float* wo  = (const float*)d_in[4];
  const float* wqi = (const float*)d_in[5];
  const float* wki = (const float*)d_in[6];
  const float* wg  = (const float*)d_in[7];
  float* out = (float*)d_out;

  float* q_all  = (float*)d_ws;                          // [S, 16*128]
  float* kmat   = q_all + (size_t)SEQL * DIMM;           // [S, 128]
  float* vmat   = kmat + (size_t)SEQL * HD;              // [S, 128]
  float* qi     = vmat + (size_t)SEQL * HD;              // [S, 4*128]
  float* ki     = qi + (size_t)SEQL * IDXH * IDXD;       // [S, 128]
  float* gate   = ki + (size_t)SEQL * IDXD;              // [S, 4]
  float* scores = gate + (size_t)SEQL * IDXH;            // [S, S]
  float* omat   = scores + (size_t)SEQL * SEQL;          // [S, 16*128]
  int* idxb     = (int*)(omat + (size_t)SEQL * DIMM);    // [S, 512]
  int* cntb     = idxb + (size_t)SEQL * KSEL;            // [S]

  const dim3 blk(256);

  gemm_wmma<<<dim3(DIMM / 128, SEQL / 128), blk, 0, stream>>>(x, wq, q_all, SEQL, DIMM, DIMM);
  gemm_wmma<<<dim3(HD / 128, SEQL / 128), blk, 0, stream>>>(x, wk, kmat, SEQL, HD, DIMM);
  gemm_wmma<<<dim3(HD / 128, SEQL / 128), blk, 0, stream>>>(x, wv, vmat, SEQL, HD, DIMM);
  gemm_wmma<<<dim3((IDXH * IDXD) / 128, SEQL / 128), blk, 0, stream>>>(x, wqi, qi, SEQL, IDXH * IDXD, DIMM);
  gemm_wmma<<<dim3(IDXD / 128, SEQL / 128), blk, 0, stream>>>(x, wki, ki, SEQL, IDXD, DIMM);

  gate_kernel<<<(SEQL * IDXH) / 8, blk, 0, stream>>>(x, wg, gate);
  rope_kernel<<<(SEQL * 17 * 32 + 255) / 256, blk, 0, stream>>>(q_all, kmat);
  fwht128<<<SEQL * IDXH, 128, 0, stream>>>(qi);
  fwht128<<<SEQL, 128, 0, stream>>>(ki);

  indexer_scores<<<dim3(SEQL / 16, SEQL / 128), blk, 0, stream>>>(qi, ki, gate, scores);
  topk_rows<<<SEQL, blk, 0, stream>>>(scores, idxb, cntb);
  attention_kernel<<<SEQL / 8, blk, 0, stream>>>(q_all, kmat, vmat, idxb, cntb, omat);

  gemm_wmma<<<dim3(DIMM / 128, SEQL / 128), blk, 0, stream>>>(omat, wo, out, SEQL, DIMM, DIMM);
}